// FCHLCuda_68289980006910
// MI455X (gfx1250) — compile-verified
//
#include <hip/hip_runtime.h>
#include <hip/hip_bf16.h>

typedef __attribute__((ext_vector_type(2))) float v2f;
typedef __attribute__((ext_vector_type(8))) float v8f;

#define NATOMSC 29
#define NSPEC   4
#define NRS2C   24
#define NRS3C   20
#define NPAIRSC 10
#define FPSZ    496      // 4*24 + 10*20*2
#define JKC     841      // 29*29
#define JKPAD   844      // 211 * 4
#define NCHUNK  211
#define PI_F    3.14159265358979f
#define W3W     12.4225798f      // sqrt(2.7/pi)*13.4
#define INV_SQRT_2PI 0.3989422804f

__global__ __launch_bounds__(256) void fchl_kernel(
    const float* __restrict__ X, const int* __restrict__ Z,
    const int* __restrict__ counts, float* __restrict__ out)
{
    __shared__ float Xs[32][3];
    __shared__ int   Zs[32];
    __shared__ float Dm[1024];     // distances, [a*32+b]
    __shared__ float Fcm[1024];    // fcut (includes validity mask)
    __shared__ float Pfm[1024];    // two-body prefactor
    __shared__ float Mum[1024];    // mu
    __shared__ float S2m[1024];    // 0.5 / sig2
    __shared__ int   PIdx[JKPAD];  // species-pair index per (j,k)
    __shared__ float Twc[JKPAD];   // 0.5*w3*cos_i
    __shared__ float Tws[JKPAD];   // 0.5*w3*sin_i
    __shared__ float Tav[JKPAD];   // 0.5*(rij+rik)
    __shared__ float Wbuf[8 * 3 * 256]; // per-wave partial WMMA tiles

    const int tid  = threadIdx.x;
    const int lane = tid & 31;
    // wave index is uniform within a wave: pin it to an SGPR so the WMMA
    // chunk loop compiles as a scalar loop (EXEC stays all-ones at v_wmma).
    const int wave = __builtin_amdgcn_readfirstlane(tid >> 5);
    const int mol  = blockIdx.x;
    const int cnt  = counts[mol];

    // ---- stage atoms ----
    if (tid < NATOMSC) {
        Xs[tid][0] = X[(mol * NATOMSC + tid) * 3 + 0];
        Xs[tid][1] = X[(mol * NATOMSC + tid) * 3 + 1];
        Xs[tid][2] = X[(mol * NATOMSC + tid) * 3 + 2];
        Zs[tid]    = Z[mol * NATOMSC + tid];
    } else if (tid < 32) {
        Xs[tid][0] = Xs[tid][1] = Xs[tid][2] = 0.f;
        Zs[tid] = 0;
    }
    __syncthreads();

    // ---- pairwise matrices ----
    for (int s = tid; s < 1024; s += 256) {
        int a = s >> 5, b = s & 31;
        float D = 1.f, f = 0.f, pf = 0.f, mu = 0.f, i2 = 1.f;
        if (a < NATOMSC && b < NATOMSC) {
            float dx = Xs[a][0] - Xs[b][0];
            float dy = Xs[a][1] - Xs[b][1];
            float dz = Xs[a][2] - Xs[b][2];
            float d2 = dx * dx + dy * dy + dz * dz;
            bool eye = (a == b);
            D = eye ? 1.f : sqrtf(fmaxf(d2, 1e-12f));
            bool nb = (!eye) && (a < cnt) && (b < cnt) && (D < 8.f);
            f = nb ? 0.5f * (__cosf(PI_F * D * 0.125f) + 1.f) : 0.f;
            float lg = __logf(D);
            float s2 = log1pf(0.32f / (D * D));
            mu = lg - 0.5f * s2;
            pf = f * __expf(-1.8f * lg) * rsqrtf(s2) * INV_SQRT_2PI;
            i2 = 0.5f / s2;
        }
        Dm[s] = D; Fcm[s] = f; Pfm[s] = pf; Mum[s] = mu; S2m[s] = i2;
    }
    for (int jk = tid; jk < JKPAD; jk += 256) {
        int p = -1;
        if (jk < JKC) {
            int j = jk / 29, k = jk - j * 29;
            int za = Zs[j], zb = Zs[k];
            int lo = za < zb ? za : zb;
            int hi = za < zb ? zb : za;
            p = lo * NSPEC - ((lo * (lo - 1)) >> 1) + (hi - lo);
        }
        PIdx[jk] = p;
    }
    __syncthreads();

    // ---- two-body ----
    for (int item = tid; item < NATOMSC * NRS2C; item += 256) {
        int i  = item / NRS2C;
        int kk = item - i * NRS2C;
        float rs  = (float)(kk + 1) * (8.f / 24.f);
        float lrs = __logf(rs);
        float inv = 1.f / rs;
        float a0 = 0.f, a1 = 0.f, a2 = 0.f, a3 = 0.f;
        int ib = i * 32;
        for (int j = 0; j < NATOMSC; ++j) {
            float t = lrs - Mum[ib + j];
            float v = Pfm[ib + j] * inv * __expf(-t * t * S2m[ib + j]);
            int e = Zs[j];
            a0 += (e == 0) ? v : 0.f;
            a1 += (e == 1) ? v : 0.f;
            a2 += (e == 2) ? v : 0.f;
            a3 += (e == 3) ? v : 0.f;
        }
        float* o = out + (size_t)(mol * NATOMSC + i) * FPSZ;
        o[0 * NRS2C + kk] = a0;
        o[1 * NRS2C + kk] = a1;
        o[2 * NRS2C + kk] = a2;
        o[3 * NRS2C + kk] = a3;
    }

    // ---- three-body: per atom i, GEMM rep3[p,col] = sum_jk P[jk,p]*Aval[jk,col] ----
    const int prow = lane & 15;
    for (int i = 0; i < NATOMSC; ++i) {
        int ib = i * 32;
        // stage 1: triplet scalars (all 256 threads)
        for (int jk = tid; jk < JKPAD; jk += 256) {
            float wc = 0.f, ws = 0.f, av = 0.f;
            if (jk < JKC) {
                int j = jk / 29, k = jk - j * 29;
                if (j != k) {
                    float fc = Fcm[ib + j] * Fcm[ib + k];
                    if (fc != 0.f) {
                        float rij = Dm[ib + j], rik = Dm[ib + k], rjk = Dm[j * 32 + k];
                        float r2ij = rij * rij, r2ik = rik * rik, r2jk = rjk * rjk;
                        float ci = fminf(fmaxf((r2ij + r2ik - r2jk) / (2.f * rij * rik), -1.f), 1.f);
                        float cj = fminf(fmaxf((r2ij + r2jk - r2ik) / (2.f * rij * rjk), -1.f), 1.f);
                        float ck = fminf(fmaxf((r2ik + r2jk - r2ij) / (2.f * rik * rjk), -1.f), 1.f);
                        float si = sqrtf(fmaxf(1.f - ci * ci, 1e-12f));
                        float ks = W3W * (1.f + 3.f * ci * cj * ck) *
                                   __expf(-0.57f * __logf(rij * rik * rjk));
                        float w3 = 0.5f * fc * ks;
                        wc = w3 * ci;
                        ws = w3 * si;
                        av = 0.5f * (rij + rik);
                    }
                }
            }
            Twc[jk] = wc; Tws[jk] = ws; Tav[jk] = av;
        }
        __syncthreads();

        // stage 2: per-wave WMMA over K chunks of 4 jk's (scalar loop, full EXEC)
        v8f c0 = {0.f,0.f,0.f,0.f,0.f,0.f,0.f,0.f};
        v8f c1 = {0.f,0.f,0.f,0.f,0.f,0.f,0.f,0.f};
        v8f c2 = {0.f,0.f,0.f,0.f,0.f,0.f,0.f,0.f};
        for (int c = wave; c < NCHUNK; c += 8) {
            int jx = c * 4 + ((lane < 16) ? 0 : 2);
            int jy = jx + 1;
            v2f a;
            a.x = (PIdx[jx] == prow) ? 1.f : 0.f;   // A[M=p, K] (16x4 layout)
            a.y = (PIdx[jy] == prow) ? 1.f : 0.f;
            float wcx = Twc[jx], wsx = Tws[jx], avx = Tav[jx];
            float wcy = Twc[jy], wsy = Tws[jy], avy = Tav[jy];
#define FCHL_TILE(ACC, T)                                                        \
            {                                                                    \
                int col = prow + 16 * (T);                                       \
                bool cA = col < NRS3C, cB = col < 2 * NRS3C;                     \
                float mm = (float)(cA ? col : col - NRS3C);                      \
                float rs = (mm + 1.f) * 0.4f;                                    \
                float wx = cA ? wcx : (cB ? wsx : 0.f);                          \
                float wy = cA ? wcy : (cB ? wsy : 0.f);                          \
                float ex = avx - rs, ey = avy - rs;                              \
                v2f b;                                                           \
                b.x = wx * __expf(-2.7f * ex * ex);                              \
                b.y = wy * __expf(-2.7f * ey * ey);                              \
                ACC = __builtin_amdgcn_wmma_f32_16x16x4_f32(                     \
                    false, a, false, b, (short)0, ACC, false, false);            \
            }
            FCHL_TILE(c0, 0)
            FCHL_TILE(c1, 1)
            FCHL_TILE(c2, 2)
#undef FCHL_TILE
        }
        // dump per-wave partial tiles to LDS (C/D layout: lane=N(+16 for M>=8), VGPR v = M%8)
        {
            int wb = wave * 768;
#pragma unroll
            for (int v = 0; v < 8; ++v) {
                Wbuf[wb + 0 * 256 + v * 32 + lane] = c0[v];
                Wbuf[wb + 1 * 256 + v * 32 + lane] = c1[v];
                Wbuf[wb + 2 * 256 + v * 32 + lane] = c2[v];
            }
        }
        __syncthreads();

        // deterministic cross-wave reduction + store (400 elements)
        float* o3 = out + (size_t)(mol * NATOMSC + i) * FPSZ + NSPEC * NRS2C;
        for (int e = tid; e < NPAIRSC * 2 * NRS3C; e += 256) {
            int p   = e / 40;
            int col = e - p * 40;
            int tile = col >> 4;
            int n    = col & 15;
            int ln   = n + ((p >= 8) ? 16 : 0);
            int v    = p & 7;
            int slot = tile * 256 + v * 32 + ln;
            float s = Wbuf[slot];
#pragma unroll
            for (int w = 1; w < 8; ++w) s += Wbuf[w * 768 + slot];
            int comp = (col < NRS3C) ? 0 : 1;
            int m    = col - comp * NRS3C;
            o3[(p * NRS3C + m) * 2 + comp] = s;
        }
        __syncthreads();
    }
}

extern "C" void kernel_launch(void* const* d_in, const int* in_sizes, int n_in,
                              void* d_out, int out_size, void* d_ws, size_t ws_size,
                              hipStream_t stream) {
    (void)n_in; (void)out_size; (void)d_ws; (void)ws_size;
    const float* X      = (const float*)d_in[0];
    const int*   Zi     = (const int*)d_in[1];
    const int*   counts = (const int*)d_in[4];
    float*       out    = (float*)d_out;
    int B = in_sizes[4];                // one workgroup per molecule
    fchl_kernel<<<B, 256, 0, stream>>>(X, Zi, counts, out);
}